// Input_embedding_73177652789808
// MI455X (gfx1250) — compile-verified
//
#include <hip/hip_runtime.h>
#include <hip/hip_bf16.h>
#include <math.h>

#define BB   8
#define NN   4096
#define CC   3
#define OO   32
#define KNB  16
#define KP1  17

typedef __attribute__((ext_vector_type(2))) float v2f;
typedef __attribute__((ext_vector_type(8))) float v8f;

// ---------------------------------------------------------------------------
// Utility: zero the stats region (deterministic across graph replays)
// ---------------------------------------------------------------------------
__global__ __launch_bounds__(256) void zero_stats_kernel(float* s, int n) {
    int i = blockIdx.x * 256 + threadIdx.x;
    if (i < n) s[i] = 0.0f;
}

// ---------------------------------------------------------------------------
// KNN: one wave (32 threads) per (batch, 16-row tile). Gram tiles via
// V_WMMA_F32_16X16X4_F32 (C=3 padded to K=4), 4 column tiles (64 columns)
// per loop iteration with A reused; top-17 per row kept in LDS.
// Row norm dropped (constant per row -> does not change per-row ordering).
// Wave-uniform prune cutoff recomputed lazily (only after insertions).
// ---------------------------------------------------------------------------
__global__ __launch_bounds__(32) void knn_kernel(const float* __restrict__ xyz,
                                                 int* __restrict__ idx_ws,
                                                 float* __restrict__ idx_out_f)
{
    __shared__ float tile[16 * 64];         // [row][col within 64-strip]
    __shared__ float ldist[16][KP1];
    __shared__ int   lind[16][KP1];

    const int lane = threadIdx.x;           // 0..31
    const int bt   = blockIdx.x;            // b*256 + rowtile
    const int b    = bt >> 8;
    const int i0   = (bt & 255) << 4;

    const float* xb = xyz + (size_t)b * CC * NN;
    const int half = lane >> 4;             // which 16-lane half
    const int m16  = lane & 15;

    // A matrix: rows i0..i0+15, K = channel (padded to 4)
    v2f a;
    if (half == 0) { a.x = xb[0 * NN + i0 + m16]; a.y = xb[1 * NN + i0 + m16]; }
    else           { a.x = xb[2 * NN + i0 + m16]; a.y = 0.0f; }

    if (lane < 16) {
        #pragma unroll
        for (int q = 0; q < KP1; ++q) { ldist[lane][q] = -3.0e38f; lind[lane][q] = 0; }
    }
    __syncthreads();

    float cut = -3.0e38f;                   // wave-uniform min of row cutoffs

    for (int j0 = 0; j0 < NN; j0 += 64) {
        // prefetch a future column strip
        __builtin_prefetch(xb + j0 + 512 + m16, 0, 0);

        float tmax = -3.0e38f;
        #pragma unroll
        for (int t = 0; t < 4; ++t) {
            const int jc = j0 + t * 16;
            // B matrix: columns jc..jc+15
            v2f bv;
            if (half == 0) { bv.x = xb[0 * NN + jc + m16]; bv.y = xb[1 * NN + jc + m16]; }
            else           { bv.x = xb[2 * NN + jc + m16]; bv.y = 0.0f; }

            // column squared norm (combine the two K-halves)
            float pn = bv.x * bv.x + bv.y * bv.y;
            float cn = pn + __shfl_xor(pn, 16);

            v8f c = {};
            c = __builtin_amdgcn_wmma_f32_16x16x4_f32(false, a, false, bv,
                                                      (short)0, c, false, false);

            // score = 2*G - |x_n|^2  (row-constant term dropped)
            #pragma unroll
            for (int r = 0; r < 8; ++r) {
                float s = 2.0f * c[r] - cn;
                tile[(r + 8 * half) * 64 + t * 16 + m16] = s;
                tmax = fmaxf(tmax, s);
            }
        }
        // wave-wide strip max -> prune test against lazy cutoff
        #pragma unroll
        for (int off = 16; off > 0; off >>= 1) tmax = fmaxf(tmax, __shfl_xor(tmax, off));
        __syncthreads();

        if (tmax > cut) {                   // wave-uniform branch
            if (lane < 16) {
                const int m = lane;
                #pragma unroll 4
                for (int n = 0; n < 64; ++n) {
                    float d = tile[m * 64 + n];
                    if (d > ldist[m][KP1 - 1]) {
                        int p = KP1 - 1;
                        while (p > 0 && ldist[m][p - 1] < d) {   // stable: ties keep lower index
                            ldist[m][p] = ldist[m][p - 1];
                            lind[m][p]  = lind[m][p - 1];
                            --p;
                        }
                        ldist[m][p] = d;
                        lind[m][p]  = j0 + n;
                    }
                }
            }
            __syncthreads();
            // recompute wave-uniform cutoff only after insertions
            float c2 = (lane < 16) ? ldist[lane][KP1 - 1] : 3.0e38f;
            #pragma unroll
            for (int off = 16; off > 0; off >>= 1) c2 = fminf(c2, __shfl_xor(c2, off));
            cut = c2;
        }
        __syncthreads();
    }

    if (lane < 16) {
        const int m = lane;
        size_t base = ((size_t)b * NN + (i0 + m)) * KNB;
        #pragma unroll
        for (int q = 0; q < KNB; ++q) {          // top-17 computed, first 16 kept
            idx_ws[base + q]    = lind[m][q];
            idx_out_f[base + q] = (float)lind[m][q];
        }
    }
}

// ---------------------------------------------------------------------------
// MLP layer 1: y0 = w0[32x3] . xyz, accumulate per-channel sum / sumsq
// ---------------------------------------------------------------------------
__global__ __launch_bounds__(256) void mlp1_kernel(const float* __restrict__ xyz,
                                                   const float* __restrict__ w0,
                                                   float* __restrict__ y0,
                                                   float* __restrict__ stats0)
{
    __shared__ float sw[OO * CC];
    __shared__ float ssum[OO], ssq[OO];
    int tid = threadIdx.x;
    if (tid < OO * CC) sw[tid] = w0[tid];
    if (tid < OO) { ssum[tid] = 0.0f; ssq[tid] = 0.0f; }
    __syncthreads();

    int p = blockIdx.x * 256 + tid;          // 0..32767
    int b = p >> 12, n = p & (NN - 1);
    const float* xb = xyz + (size_t)b * CC * NN + n;
    float x0 = xb[0], x1 = xb[NN], x2 = xb[2 * NN];
    float* yb = y0 + (size_t)b * OO * NN + n;
    #pragma unroll
    for (int o = 0; o < OO; ++o) {
        float y = sw[o * 3] * x0 + sw[o * 3 + 1] * x1 + sw[o * 3 + 2] * x2;
        yb[(size_t)o * NN] = y;
        atomicAdd(&ssum[o], y);
        atomicAdd(&ssq[o], y * y);
    }
    __syncthreads();
    if (tid < OO) {
        atomicAdd(&stats0[tid], ssum[tid]);
        atomicAdd(&stats0[OO + tid], ssq[tid]);
    }
}

// ---------------------------------------------------------------------------
// BN (train-mode batch stats) + LeakyReLU(0.1), elementwise
// ---------------------------------------------------------------------------
__global__ __launch_bounds__(256) void bnact_kernel(const float* __restrict__ y,
                                                    const float* __restrict__ stats,
                                                    const float* __restrict__ g,
                                                    const float* __restrict__ bt,
                                                    float* __restrict__ out,
                                                    int out_ch_per_b)
{
    size_t i = (size_t)blockIdx.x * 256 + threadIdx.x;    // < 8*32*4096
    int n = (int)(i & (NN - 1));
    int o = (int)((i >> 12) & (OO - 1));
    int b = (int)(i >> 17);
    const float cnt = (float)(BB * NN);                   // mean over (B, N)
    float s = stats[o], s2 = stats[OO + o];
    float mean = s / cnt;
    float var  = fmaxf(s2 / cnt - mean * mean, 0.0f);     // biased, as torch BN
    float sc   = g[o] * rsqrtf(var + 1e-5f);
    float v = (y[i] - mean) * sc + bt[o];
    v = (v >= 0.0f) ? v : 0.1f * v;
    out[((size_t)b * out_ch_per_b + o) * NN + n] = v;
}

// ---------------------------------------------------------------------------
// MLP layer 2: y1 = w1[32x32] . x0, accumulate per-channel sum / sumsq
// ---------------------------------------------------------------------------
__global__ __launch_bounds__(256) void mlp2_kernel(const float* __restrict__ x0,
                                                   const float* __restrict__ w1,
                                                   float* __restrict__ y1,
                                                   float* __restrict__ stats1)
{
    __shared__ float sw[OO * OO];
    __shared__ float ssum[OO], ssq[OO];
    int tid = threadIdx.x;
    for (int i = tid; i < OO * OO; i += 256) sw[i] = w1[i];
    if (tid < OO) { ssum[tid] = 0.0f; ssq[tid] = 0.0f; }
    __syncthreads();

    int p = blockIdx.x * 256 + tid;
    int b = p >> 12, n = p & (NN - 1);
    const float* xb = x0 + (size_t)b * OO * NN + n;
    float xin[OO];
    #pragma unroll
    for (int c = 0; c < OO; ++c) xin[c] = xb[(size_t)c * NN];
    float* yb = y1 + (size_t)b * OO * NN + n;
    #pragma unroll
    for (int o = 0; o < OO; ++o) {
        float y = 0.0f;
        #pragma unroll
        for (int c = 0; c < OO; ++c) y += sw[o * OO + c] * xin[c];
        yb[(size_t)o * NN] = y;
        atomicAdd(&ssum[o], y);
        atomicAdd(&ssq[o], y * y);
    }
    __syncthreads();
    if (tid < OO) {
        atomicAdd(&stats1[tid], ssum[tid]);
        atomicAdd(&stats1[OO + tid], ssq[tid]);
    }
}

// ---------------------------------------------------------------------------
// Neighbor-feature tiles: res[k][o] = (nbr - center) . w_nn^T via two
// V_WMMA_F32_16X16X4_F32 (O split into two 16-wide tiles, K = C padded to 4)
// ---------------------------------------------------------------------------
__device__ inline void nn_compute(const float* __restrict__ xyz,
                                  const float* __restrict__ w_nn,
                                  const int* __restrict__ iidx,
                                  int b, int n, int lane, v8f& r0, v8f& r1)
{
    const int half = lane >> 4, m16 = lane & 15;
    const float* xb = xyz + (size_t)b * CC * NN;
    int nb = iidx[((size_t)b * NN + n) * KNB + m16];

    v2f a;
    if (half == 0) {
        a.x = xb[0 * NN + nb] - xb[0 * NN + n];
        a.y = xb[1 * NN + nb] - xb[1 * NN + n];
    } else {
        a.x = xb[2 * NN + nb] - xb[2 * NN + n];
        a.y = 0.0f;
    }
    v2f b0, b1;
    if (half == 0) {
        b0.x = w_nn[m16 * 3 + 0];        b0.y = w_nn[m16 * 3 + 1];
        b1.x = w_nn[(16 + m16) * 3 + 0]; b1.y = w_nn[(16 + m16) * 3 + 1];
    } else {
        b0.x = w_nn[m16 * 3 + 2];        b0.y = 0.0f;
        b1.x = w_nn[(16 + m16) * 3 + 2]; b1.y = 0.0f;
    }
    v8f z0 = {}, z1 = {};
    r0 = __builtin_amdgcn_wmma_f32_16x16x4_f32(false, a, false, b0, (short)0, z0, false, false);
    r1 = __builtin_amdgcn_wmma_f32_16x16x4_f32(false, a, false, b1, (short)0, z1, false, false);
}

// Pass 1: per-batch sum / sumsq of res (for unbiased std)
__global__ __launch_bounds__(256) void nn_stats_kernel(const float* __restrict__ xyz,
                                                       const float* __restrict__ w_nn,
                                                       const int* __restrict__ iidx,
                                                       float* __restrict__ statsnn)
{
    int lane = threadIdx.x & 31;
    int widx = threadIdx.x >> 5;
    int p = blockIdx.x * 8 + widx;           // one wave per point
    int b = p >> 12, n = p & (NN - 1);

    v8f r0, r1;
    nn_compute(xyz, w_nn, iidx, b, n, lane, r0, r1);

    float s = 0.0f, s2 = 0.0f;
    #pragma unroll
    for (int r = 0; r < 8; ++r) {
        s  += r0[r] + r1[r];
        s2 += r0[r] * r0[r] + r1[r] * r1[r];
    }
    #pragma unroll
    for (int off = 16; off > 0; off >>= 1) {
        s  += __shfl_xor(s, off);
        s2 += __shfl_xor(s2, off);
    }
    if (lane == 0) {
        atomicAdd(&statsnn[b], s);
        atomicAdd(&statsnn[BB + b], s2);
    }
}

// Pass 2: recompute res, normalize by per-batch std, affine, max over k
__global__ __launch_bounds__(256) void nn_final_kernel(const float* __restrict__ xyz,
                                                       const float* __restrict__ w_nn,
                                                       const int* __restrict__ iidx,
                                                       const float* __restrict__ statsnn,
                                                       const float* __restrict__ alpha,
                                                       const float* __restrict__ beta_a,
                                                       float* __restrict__ out)
{
    int lane = threadIdx.x & 31;
    int widx = threadIdx.x >> 5;
    int p = blockIdx.x * 8 + widx;
    int b = p >> 12, n = p & (NN - 1);

    v8f r0, r1;
    nn_compute(xyz, w_nn, iidx, b, n, lane, r0, r1);

    const float cnt = (float)((size_t)NN * KNB * OO);     // elements per batch
    float s  = statsnn[b];
    float s2 = statsnn[BB + b];
    float mean = s / cnt;
    float var  = fmaxf((s2 - s * mean) / (cnt - 1.0f), 0.0f);   // unbiased (torch.std)
    float inv  = 1.0f / (sqrtf(var) + 1e-5f);

    const int m16 = lane & 15;
    float a0 = alpha[m16],      be0 = beta_a[m16];
    float a1 = alpha[16 + m16], be1 = beta_a[16 + m16];

    float m0 = -3.0e38f, m1 = -3.0e38f;
    #pragma unroll
    for (int r = 0; r < 8; ++r) {
        m0 = fmaxf(m0, a0 * (r0[r] * inv) + be0);
        m1 = fmaxf(m1, a1 * (r1[r] * inv) + be1);
    }
    m0 = fmaxf(m0, __shfl_xor(m0, 16));   // combine the two row-halves
    m1 = fmaxf(m1, __shfl_xor(m1, 16));

    if (lane < 16) {
        out[((size_t)b * 64 + 32 + m16) * NN + n] = m0;
        out[((size_t)b * 64 + 48 + m16) * NN + n] = m1;
    }
}

// ---------------------------------------------------------------------------
extern "C" void kernel_launch(void* const* d_in, const int* in_sizes, int n_in,
                              void* d_out, int out_size, void* d_ws, size_t ws_size,
                              hipStream_t stream)
{
    const float* xyz    = (const float*)d_in[0];
    const float* w_nn   = (const float*)d_in[1];
    const float* alpha  = (const float*)d_in[2];
    const float* beta_a = (const float*)d_in[3];
    const float* w0     = (const float*)d_in[4];
    const float* g0     = (const float*)d_in[5];
    const float* b0     = (const float*)d_in[6];
    const float* w1     = (const float*)d_in[7];
    const float* g1     = (const float*)d_in[8];
    const float* b1     = (const float*)d_in[9];
    // d_in[10] = k (16) — fixed problem size, baked in as KNB

    float* pf   = (float*)d_out;                          // [8,64,4096]
    float* idxf = pf + (size_t)BB * 64 * NN;              // [8,4096,16] as floats

    // workspace layout (floats)
    int*   iidx = (int*)d_ws;                             // 524288 ints
    float* W    = (float*)d_ws;
    float* y0v  = W + 524288;                             // 8*32*4096
    float* x0v  = y0v + 1048576;
    float* y1v  = x0v + 1048576;
    float* st0  = y1v + 1048576;                          // 64
    float* st1  = st0 + 64;                               // 64
    float* stn  = st1 + 64;                               // 16

    // 1) zero the 144 stat floats (contiguous from st0)
    zero_stats_kernel<<<1, 256, 0, stream>>>(st0, 144);

    // 2) KNN via f32 WMMA Gram tiles + LDS top-17
    knn_kernel<<<BB * (NN / 16), 32, 0, stream>>>(xyz, iidx, idxf);

    // 3) MLP branch
    mlp1_kernel<<<(BB * NN) / 256, 256, 0, stream>>>(xyz, w0, y0v, st0);
    bnact_kernel<<<(BB * OO * NN) / 256, 256, 0, stream>>>(y0v, st0, g0, b0, x0v, OO);
    mlp2_kernel<<<(BB * NN) / 256, 256, 0, stream>>>(x0v, w1, y1v, st1);
    bnact_kernel<<<(BB * OO * NN) / 256, 256, 0, stream>>>(y1v, st1, g1, b1, pf, 64);

    // 4) neighbor branch (WMMA res tiles): stats pass, then normalize+max pass
    nn_stats_kernel<<<(BB * NN) / 8, 256, 0, stream>>>(xyz, w_nn, iidx, stn);
    nn_final_kernel<<<(BB * NN) / 8, 256, 0, stream>>>(xyz, w_nn, iidx, stn,
                                                       alpha, beta_a, pf);
    (void)in_sizes; (void)n_in; (void)out_size; (void)ws_size;
}